// GCN_1254130450622
// MI455X (gfx1250) — compile-verified
//
#include <hip/hip_runtime.h>

#define HID 128
#define NGRAPH 512

typedef float v2f __attribute__((ext_vector_type(2)));
typedef float v8f __attribute__((ext_vector_type(8)));

// ---------------------------------------------------------------------------
// 1) Node feature build: x[i] = concat(z_emb[z[i]], node_emb[node_id[i]]) [N,256]
//    One float4 per thread (64 per node).
// ---------------------------------------------------------------------------
__global__ void embed_kernel(const int* __restrict__ z, const int* __restrict__ nid,
                             const float* __restrict__ z_emb, const float* __restrict__ n_emb,
                             float* __restrict__ x, int n) {
  long t = (long)blockIdx.x * blockDim.x + threadIdx.x;
  int i = (int)(t >> 6);
  int c = (int)(t & 63);
  if (i >= n) return;
  float4 v;
  if (c < 32) {
    v = ((const float4*)(z_emb + (long)z[i] * HID))[c];
  } else {
    v = ((const float4*)(n_emb + (long)nid[i] * HID))[c - 32];
  }
  ((float4*)(x + (long)i * 256))[c] = v;
}

// ---------------------------------------------------------------------------
// 2) Edge scatter: agg[dst] += x[src]   (float4 gather + 4x global f32 atomics)
//    Thread t handles (edge = t / (KW/4), float4 chunk = t % (KW/4)).
// ---------------------------------------------------------------------------
template <int KW>
__global__ void scatter_add(const float* __restrict__ x, const int* __restrict__ src,
                            const int* __restrict__ dst, float* __restrict__ agg, long nwork) {
  const int C4 = KW / 4;
  long t = (long)blockIdx.x * blockDim.x + threadIdx.x;
  if (t >= nwork) return;
  long e = t / C4;
  int c = (int)(t % C4);
  int s = src[e], d = dst[e];
  float4 v = ((const float4*)(x + (long)s * KW))[c];
  float* p = agg + (long)d * KW + (long)c * 4;
  atomicAdd(p + 0, v.x);
  atomicAdd(p + 1, v.y);
  atomicAdd(p + 2, v.z);
  atomicAdd(p + 3, v.w);
}

// ---------------------------------------------------------------------------
// 3) GIN linear via V_WMMA_F32_16X16X4_F32 (fp32-exact matrix path).
//    Block = 256 threads (8 waves). Block owns 64 output rows (4 M-tiles);
//    wave w owns output columns [16w, 16w+16). The A-tile (x+agg fused) is
//    staged in LDS; each B fragment (2 W elements/lane) is reused across the
//    4 M-tiles -> 4 WMMAs per B load, 4x less L2 weight traffic.
//    ISA layouts (CDNA5 ISA):
//      A 16x4 f32: lane&15 = M; VGPR0 holds K=0 (lanes 0-15) / K=2 (16-31),
//                  VGPR1 holds K=1 / K=3.
//      B 4x16 f32: symmetric — lane&15 = N, lane>>4 selects K pair.
//      C 16x16 f32: VGPR v -> row v + 8*(lane>>4), col = lane&15.
// ---------------------------------------------------------------------------
#define MT 4   // M-tiles per block (64 rows)

template <int KIN, bool RELU>
__global__ void __launch_bounds__(256) gin_gemm(const float* __restrict__ x,
                                                const float* __restrict__ agg,
                                                const float* __restrict__ W,   // [KIN, HID]
                                                const float* __restrict__ bias,
                                                float* __restrict__ out,       // [n, HID]
                                                int nrows) {
  __shared__ __align__(16) float At[16 * MT][KIN];
  const int tid = threadIdx.x;
  const int rowBase = blockIdx.x * (16 * MT);

  // Cooperative A-tile load, fusing h = x + agg. Clamp tail rows (row m of A
  // only affects row m of C; clamped rows are never stored).
  for (int idx = tid * 4; idx < 16 * MT * KIN; idx += 256 * 4) {
    int r = idx / KIN, k = idx % KIN;
    int gr = rowBase + r;
    if (gr >= nrows) gr = nrows - 1;
    float4 xv = *(const float4*)(x + (long)gr * KIN + k);
    float4 av = *(const float4*)(agg + (long)gr * KIN + k);
    *(float4*)(&At[r][k]) = make_float4(xv.x + av.x, xv.y + av.y, xv.z + av.z, xv.w + av.w);
  }
  __syncthreads();

  const int wave  = tid >> 5;          // 0..7 -> N tile
  const int lane  = tid & 31;
  const int mrow  = lane & 15;         // A row-in-tile / B,C column-in-tile
  const int khalf = lane >> 4;         // 0: K pair {0,1}, 1: K pair {2,3}
  const int col   = wave * 16 + mrow;  // global output column

  v8f acc[MT];
#pragma unroll
  for (int mt = 0; mt < MT; ++mt) acc[mt] = (v8f){};

  for (int k = 0; k < KIN; k += 4) {
    v2f b;
    b.x = W[(long)(k + 2 * khalf) * HID + col];
    b.y = W[(long)(k + 2 * khalf + 1) * HID + col];
#pragma unroll
    for (int mt = 0; mt < MT; ++mt) {
      v2f a;
      a.x = At[mt * 16 + mrow][k + 2 * khalf];
      a.y = At[mt * 16 + mrow][k + 2 * khalf + 1];
      acc[mt] = __builtin_amdgcn_wmma_f32_16x16x4_f32(false, a, false, b, (short)0,
                                                      acc[mt], false, false);
    }
  }

  const float bcol = bias[col];
  if (rowBase + 16 * MT <= nrows) {
    // Fast path: full tile, unconditional stores (no EXEC churn).
#pragma unroll
    for (int mt = 0; mt < MT; ++mt) {
#pragma unroll
      for (int v = 0; v < 8; ++v) {
        int gr = rowBase + mt * 16 + v + 8 * khalf;
        float r = acc[mt][v] + bcol;
        if (RELU) r = fmaxf(r, 0.0f);
        out[(long)gr * HID + col] = r;
      }
    }
  } else {
#pragma unroll
    for (int mt = 0; mt < MT; ++mt) {
#pragma unroll
      for (int v = 0; v < 8; ++v) {
        int gr = rowBase + mt * 16 + v + 8 * khalf;
        float r = acc[mt][v] + bcol;
        if (RELU) r = fmaxf(r, 0.0f);
        if (gr < nrows) out[(long)gr * HID + col] = r;
      }
    }
  }
}

// ---------------------------------------------------------------------------
// 4) Global mean-pool scatter: sums[batch[i]] += x[i]; counts[batch[i]] += 1
// ---------------------------------------------------------------------------
__global__ void pool_scatter(const float* __restrict__ x, const int* __restrict__ batch,
                             float* __restrict__ sums, float* __restrict__ counts, int n) {
  long t = (long)blockIdx.x * blockDim.x + threadIdx.x;
  int i = (int)(t >> 5);
  int c = (int)(t & 31);
  if (i >= n) return;
  int g = batch[i];
  float4 v = ((const float4*)(x + (long)i * HID))[c];
  float* p = sums + (long)g * HID + (long)c * 4;
  atomicAdd(p + 0, v.x);
  atomicAdd(p + 1, v.y);
  atomicAdd(p + 2, v.z);
  atomicAdd(p + 3, v.w);
  if (c == 0) atomicAdd(counts + g, 1.0f);
}

// ---------------------------------------------------------------------------
// 5) Head MLP per graph: pooled = sums/max(cnt,1); Linear->BN(eval)->ReLU->Linear
//    Block = 128 threads = one graph.
// ---------------------------------------------------------------------------
__global__ void __launch_bounds__(128) mlp_kernel(const float* __restrict__ sums,
                                                  const float* __restrict__ counts,
                                                  const float* __restrict__ W1,   // [HID, HID]
                                                  const float* __restrict__ b1,
                                                  const float* __restrict__ gamma,
                                                  const float* __restrict__ beta,
                                                  const float* __restrict__ W2,   // [HID, 2]
                                                  const float* __restrict__ b2,
                                                  float* __restrict__ out) {      // [G, 2]
  __shared__ float prow[HID];
  __shared__ float part0[HID];
  __shared__ float part1[HID];
  const int g = blockIdx.x;
  const int j = threadIdx.x;

  float cnt = fmaxf(counts[g], 1.0f);
  prow[j] = sums[(long)g * HID + j] / cnt;
  __syncthreads();

  float acc = b1[j];
  for (int k = 0; k < HID; ++k) acc += prow[k] * W1[(long)k * HID + j];
  float scale = gamma[j] * rsqrtf(1.0f + 1e-5f);
  acc = fmaxf(acc * scale + beta[j], 0.0f);

  part0[j] = acc * W2[j * 2 + 0];
  part1[j] = acc * W2[j * 2 + 1];
  __syncthreads();
  for (int s = 64; s > 0; s >>= 1) {
    if (j < s) { part0[j] += part0[j + s]; part1[j] += part1[j + s]; }
    __syncthreads();
  }
  if (j == 0) {
    out[g * 2 + 0] = part0[0] + b2[0];
    out[g * 2 + 1] = part1[0] + b2[1];
  }
}

// ---------------------------------------------------------------------------
extern "C" void kernel_launch(void* const* d_in, const int* in_sizes, int n_in,
                              void* d_out, int out_size, void* d_ws, size_t ws_size,
                              hipStream_t stream) {
  const int*   z      = (const int*)d_in[1];
  const int*   ei     = (const int*)d_in[2];
  const int*   batch  = (const int*)d_in[3];
  const int*   nid    = (const int*)d_in[4];
  const float* z_emb  = (const float*)d_in[5];
  const float* n_emb  = (const float*)d_in[6];
  const float* W0     = (const float*)d_in[7];
  const float* b0     = (const float*)d_in[8];
  const float* W1     = (const float*)d_in[9];
  const float* b1     = (const float*)d_in[10];
  const float* W2     = (const float*)d_in[11];
  const float* b2     = (const float*)d_in[12];
  const float* mW1    = (const float*)d_in[13];
  const float* mb1    = (const float*)d_in[14];
  const float* gamma  = (const float*)d_in[15];
  const float* beta   = (const float*)d_in[16];
  const float* mW2    = (const float*)d_in[17];
  const float* mb2    = (const float*)d_in[18];

  const int N = in_sizes[1];        // nodes
  const int E = in_sizes[2] / 2;    // edges
  const int* src = ei;
  const int* dst = ei + E;

  // Workspace layout (floats)
  float* x0     = (float*)d_ws;                 // [N, 256]
  float* agg    = x0  + (size_t)N * 256;        // [N, 256] (reused at width 128)
  float* h1     = agg + (size_t)N * 256;        // [N, 128]
  float* h2     = h1  + (size_t)N * 128;        // [N, 128]
  float* sums   = h2  + (size_t)N * 128;        // [512, 128]
  float* counts = sums + (size_t)NGRAPH * HID;  // [512]

  const unsigned gemmBlocks = (unsigned)((N + 16 * MT - 1) / (16 * MT));

  // ---- embed
  {
    long nt = (long)N * 64;
    embed_kernel<<<(unsigned)((nt + 255) / 256), 256, 0, stream>>>(z, nid, z_emb, n_emb, x0, N);
  }

  // ---- conv0 (width 256 in, ReLU)
  hipMemsetAsync(agg, 0, (size_t)N * 256 * sizeof(float), stream);
  {
    long nw = (long)E * 64;
    scatter_add<256><<<(unsigned)((nw + 255) / 256), 256, 0, stream>>>(x0, src, dst, agg, nw);
  }
  gin_gemm<256, true><<<gemmBlocks, 256, 0, stream>>>(x0, agg, W0, b0, h1, N);

  // ---- conv1 (width 128, ReLU)
  hipMemsetAsync(agg, 0, (size_t)N * 128 * sizeof(float), stream);
  {
    long nw = (long)E * 32;
    scatter_add<128><<<(unsigned)((nw + 255) / 256), 256, 0, stream>>>(h1, src, dst, agg, nw);
  }
  gin_gemm<128, true><<<gemmBlocks, 256, 0, stream>>>(h1, agg, W1, b1, h2, N);

  // ---- conv2 (width 128, no ReLU)
  hipMemsetAsync(agg, 0, (size_t)N * 128 * sizeof(float), stream);
  {
    long nw = (long)E * 32;
    scatter_add<128><<<(unsigned)((nw + 255) / 256), 256, 0, stream>>>(h2, src, dst, agg, nw);
  }
  gin_gemm<128, false><<<gemmBlocks, 256, 0, stream>>>(h2, agg, W2, b2, h1, N);

  // ---- mean pool
  hipMemsetAsync(sums, 0, (size_t)(NGRAPH * HID + NGRAPH) * sizeof(float), stream);
  {
    long nw = (long)N * 32;
    pool_scatter<<<(unsigned)((nw + 255) / 256), 256, 0, stream>>>(h1, batch, sums, counts, N);
  }

  // ---- head MLP
  mlp_kernel<<<NGRAPH, 128, 0, stream>>>(sums, counts, mW1, mb1, gamma, beta, mW2, mb2,
                                         (float*)d_out);
}